// GNNActor_6571299963316
// MI455X (gfx1250) — compile-verified
//
#include <hip/hip_runtime.h>
#include <hip/hip_bf16.h>

typedef __attribute__((ext_vector_type(2))) float v2f;
typedef __attribute__((ext_vector_type(8))) float v8f;

// ---------------------------------------------------------------------------
// Kernel 1: deg[i] = 1.0 (self loop contribution)
// ---------------------------------------------------------------------------
__global__ void k_deg_init(float* __restrict__ deg, int N) {
  int i = blockIdx.x * blockDim.x + threadIdx.x;
  if (i < N) deg[i] = 1.0f;
}

// ---------------------------------------------------------------------------
// Kernel 2: deg[col[e]] += 1 for every edge
// ---------------------------------------------------------------------------
__global__ void k_deg_accum(const long long* __restrict__ ei, float* __restrict__ deg, int E) {
  int e = blockIdx.x * blockDim.x + threadIdx.x;
  if (e < E) {
    int col = (int)ei[(long long)E + e];
    atomicAdd(deg + col, 1.0f);
  }
}

// ---------------------------------------------------------------------------
// Kernel 3: dinv[i] = rsqrt(deg[i])   (deg >= 1 always, so no zero guard)
// ---------------------------------------------------------------------------
__global__ void k_dinv(const float* __restrict__ deg, float* __restrict__ dinv, int N) {
  int i = blockIdx.x * blockDim.x + threadIdx.x;
  if (i < N) dinv[i] = rsqrtf(deg[i]);
}

// ---------------------------------------------------------------------------
// Kernel 4: xw = x @ W   [N x 64] @ [64 x 64], fp32 WMMA 16x16x4.
// One wave owns a 16-row slab and all 4 column tiles (16 cols each).
// A fragment (16x4 f32, 2 VGPR): lane l -> M = l%16, K = 2*(l/16) + vgpr
// B fragment (4x16 f32, 2 VGPR): lane l -> N = l%16, K = 2*(l/16) + vgpr
// C/D (16x16 f32, 8 VGPR):       lane l, vgpr v -> N = l%16, M = v + 8*(l/16)
// ---------------------------------------------------------------------------
__global__ void k_gemm_xw(const float* __restrict__ x, const float* __restrict__ W,
                          float* __restrict__ xw, int N) {
  const int lane   = threadIdx.x & 31;
  const int waveId = blockIdx.x * (blockDim.x >> 5) + (threadIdx.x >> 5);
  const int row0   = waveId * 16;
  if (row0 >= N) return;                    // wave-uniform: EXEC stays all-ones
  const int m    = lane & 15;
  const int half = lane >> 4;               // 0 or 1

  v8f acc0 = {}, acc1 = {}, acc2 = {}, acc3 = {};
  const float* arow = x + (long long)(row0 + m) * 64;

#pragma unroll
  for (int k0 = 0; k0 < 64; k0 += 4) {
    const int ka = k0 + 2 * half;           // K index of vgpr0 for this lane
    v2f a;  a.x = arow[ka];  a.y = arow[ka + 1];

    const float* wr = W + ka * 64;          // row K=ka of W (row-major 64x64)
    v2f b0; b0.x = wr[      m]; b0.y = wr[64 +      m];
    v2f b1; b1.x = wr[16 +  m]; b1.y = wr[64 + 16 + m];
    v2f b2; b2.x = wr[32 +  m]; b2.y = wr[64 + 32 + m];
    v2f b3; b3.x = wr[48 +  m]; b3.y = wr[64 + 48 + m];

    acc0 = __builtin_amdgcn_wmma_f32_16x16x4_f32(false, a, false, b0, (short)0, acc0, false, false);
    acc1 = __builtin_amdgcn_wmma_f32_16x16x4_f32(false, a, false, b1, (short)0, acc1, false, false);
    acc2 = __builtin_amdgcn_wmma_f32_16x16x4_f32(false, a, false, b2, (short)0, acc2, false, false);
    acc3 = __builtin_amdgcn_wmma_f32_16x16x4_f32(false, a, false, b3, (short)0, acc3, false, false);
  }

#pragma unroll
  for (int v = 0; v < 8; ++v) {
    const int r = row0 + v + 8 * half;
    float* orow = xw + (long long)r * 64;
    orow[     m] = acc0[v];
    orow[16 + m] = acc1[v];
    orow[32 + m] = acc2[v];
    orow[48 + m] = acc3[v];
  }
}

// ---------------------------------------------------------------------------
// Kernel 5: self-loop term  agg[i,:] = xw[i,:] * dinv[i]^2
// ---------------------------------------------------------------------------
__global__ void k_agg_init(const float* __restrict__ xw, const float* __restrict__ dinv,
                           float* __restrict__ agg, int total) {
  int t = blockIdx.x * blockDim.x + threadIdx.x;
  if (t < total) {
    int i = t >> 6;
    float d = dinv[i];
    agg[t] = xw[t] * d * d;
  }
}

// ---------------------------------------------------------------------------
// Kernel 6: edge scatter. 16 threads per edge, float4 gather + 4 f32 atomics.
// agg[col,:] += xw[row,:] * dinv[row]*dinv[col]
// ---------------------------------------------------------------------------
__global__ void k_scatter(const long long* __restrict__ ei, const float* __restrict__ xw,
                          const float* __restrict__ dinv, float* __restrict__ agg, int E) {
  long long tid = (long long)blockIdx.x * blockDim.x + threadIdx.x;
  int e     = (int)(tid >> 4);
  int chunk = (int)(tid & 15);
  if (e >= E) return;
  int row = (int)ei[e];
  int col = (int)ei[(long long)E + e];
  float norm = dinv[row] * dinv[col];
  const float4 v = *reinterpret_cast<const float4*>(xw + (long long)row * 64 + chunk * 4);
  float* dst = agg + (long long)col * 64 + chunk * 4;
  atomicAdd(dst + 0, v.x * norm);
  atomicAdd(dst + 1, v.y * norm);
  atomicAdd(dst + 2, v.z * norm);
  atomicAdd(dst + 3, v.w * norm);
}

// ---------------------------------------------------------------------------
// Kernel 7: fused epilogue.  One wave per node (8 nodes / 256-thread block).
// h0 = relu(agg + b_gcn) + x ; h1 = relu(h0@w1+b1) ; h2 = relu(h1@w2+b2)
// out = h2@w3 + b3
// ---------------------------------------------------------------------------
__global__ void k_mlp(const float* __restrict__ agg, const float* __restrict__ x,
                      const float* __restrict__ bg,
                      const float* __restrict__ w1, const float* __restrict__ b1,
                      const float* __restrict__ w2, const float* __restrict__ b2,
                      const float* __restrict__ w3, const float* __restrict__ b3,
                      float* __restrict__ out, int N) {
  __shared__ float sh0[8][64];
  __shared__ float sh1[8][32];
  __shared__ float sh2[8][32];
  const int lane = threadIdx.x & 31;
  const int wv   = threadIdx.x >> 5;
  const int n    = blockIdx.x * 8 + wv;
  if (n >= N) return;                        // terminated waves don't block barrier
  const long long base = (long long)n * 64;

#pragma unroll
  for (int c = lane; c < 64; c += 32) {
    float v = agg[base + c] + bg[c];
    v = v > 0.0f ? v : 0.0f;                 // relu
    sh0[wv][c] = v + x[base + c];            // residual
  }
  __syncthreads();

  float t1 = b1[lane];
#pragma unroll 8
  for (int k = 0; k < 64; ++k) t1 = fmaf(sh0[wv][k], w1[k * 32 + lane], t1);
  sh1[wv][lane] = t1 > 0.0f ? t1 : 0.0f;
  __syncthreads();

  float t2 = b2[lane];
#pragma unroll 8
  for (int k = 0; k < 32; ++k) t2 = fmaf(sh1[wv][k], w2[k * 32 + lane], t2);
  sh2[wv][lane] = t2 > 0.0f ? t2 : 0.0f;
  __syncthreads();

  if (lane < 4) {
    float t3 = b3[lane];
#pragma unroll 8
    for (int k = 0; k < 32; ++k) t3 = fmaf(sh2[wv][k], w3[k * 4 + lane], t3);
    out[(long long)n * 4 + lane] = t3;
  }
}

// ---------------------------------------------------------------------------
extern "C" void kernel_launch(void* const* d_in, const int* in_sizes, int n_in,
                              void* d_out, int out_size, void* d_ws, size_t ws_size,
                              hipStream_t stream) {
  const float*     x  = (const float*)d_in[0];
  const long long* ei = (const long long*)d_in[1];   // int64 edge_index [2,E]
  const float*     Wg = (const float*)d_in[2];
  const float*     bg = (const float*)d_in[3];
  const float*     w1 = (const float*)d_in[4];
  const float*     b1 = (const float*)d_in[5];
  const float*     w2 = (const float*)d_in[6];
  const float*     b2 = (const float*)d_in[7];
  const float*     w3 = (const float*)d_in[8];
  const float*     b3 = (const float*)d_in[9];
  float* out = (float*)d_out;

  const int N = in_sizes[0] / 64;
  const int E = in_sizes[1] / 2;

  // Workspace layout: xw[N*64] | agg[N*64] | deg[N] | dinv[N]  (~52 MB)
  float* ws   = (float*)d_ws;
  float* xw   = ws;
  float* agg  = xw  + (size_t)N * 64;
  float* deg  = agg + (size_t)N * 64;
  float* dinv = deg + N;

  const int T = 256;
  // 1) dense GEMM (WMMA): one wave per 16 rows, 8 waves per block
  const int waves  = (N + 15) / 16;
  k_gemm_xw<<<(waves + 7) / 8, T, 0, stream>>>(x, Wg, xw, N);
  // 2-4) degrees and normalization
  k_deg_init <<<(N + T - 1) / T, T, 0, stream>>>(deg, N);
  k_deg_accum<<<(E + T - 1) / T, T, 0, stream>>>(ei, deg, E);
  k_dinv     <<<(N + T - 1) / T, T, 0, stream>>>(deg, dinv, N);
  // 5) self-loop term into accumulator
  const int total = N * 64;
  k_agg_init <<<(total + T - 1) / T, T, 0, stream>>>(xw, dinv, agg, total);
  // 6) edge scatter (16 threads / edge)
  const long long st = (long long)E * 16;
  k_scatter  <<<(int)((st + T - 1) / T), T, 0, stream>>>(ei, xw, dinv, agg, E);
  // 7) fused relu/residual + MLP
  k_mlp      <<<(N + 7) / 8, T, 0, stream>>>(agg, x, bg, w1, b1, w2, b2, w3, b3, out, N);
}